// Aaren_39505109188891
// MI455X (gfx1250) — compile-verified
//
#include <hip/hip_runtime.h>

// Aaren single-query online-softmax prefix attention, MI455X (gfx1250).
// L=8192 tokens, D=2048 channels, fp32 throughout (matches reference).

constexpr int LTOK = 8192;   // L
constexpr int DCH  = 2048;   // D
constexpr int GRP  = 64;     // row groups
constexpr int RPG  = 128;    // rows per group (GRP*RPG == LTOK)
constexpr int NCHK = LTOK / 16;  // 512 16-row chunks

typedef __attribute__((ext_vector_type(2))) float v2f;
typedef __attribute__((ext_vector_type(8))) float v8f;

#define NEG_INF (-__builtin_inff())

__device__ __forceinline__ void lse_comb(float& m, float& u, float m2, float u2) {
  float mf = fmaxf(m, m2);
  u = u * __expf(m - mf) + u2 * __expf(m2 - mf);
  m = mf;
}

// ---------------------------------------------------------------------------
// Kernel 1: s[t] = <K[t,:], q>.  One wave per row, float4 coalesced loads.
// ---------------------------------------------------------------------------
__global__ void aaren_logits(const float* __restrict__ K,
                             const float* __restrict__ q,
                             float* __restrict__ s) {
  const int lane = threadIdx.x & 31;
  const int wave = threadIdx.x >> 5;
  const int row  = blockIdx.x * 8 + wave;
  const float4* K4 = reinterpret_cast<const float4*>(K + (size_t)row * DCH);
  const float4* q4 = reinterpret_cast<const float4*>(q);
  float acc = 0.0f;
#pragma unroll 4
  for (int i = lane; i < DCH / 4; i += 32) {
    float4 k = K4[i];
    float4 x = q4[i];
    acc += k.x * x.x + k.y * x.y + k.z * x.z + k.w * x.w;
  }
#pragma unroll
  for (int off = 16; off > 0; off >>= 1)
    acc += __shfl_down(acc, off, 32);
  if (lane == 0) s[row] = acc;
}

// ---------------------------------------------------------------------------
// Kernel 2: single-block scan of the (max, normalizer) monoid over L scalars.
// Outputs M[t] (running max), U[t] (normalizer), invU[t] = 1/U[t].
// ---------------------------------------------------------------------------
__global__ void aaren_scan(const float* __restrict__ s,
                           float* __restrict__ Mout,
                           float* __restrict__ Uout,
                           float* __restrict__ iU) {
  const int tid  = threadIdx.x;
  const int lane = tid & 31;
  const int wave = tid >> 5;
  const int base = tid * 32;

  float m = NEG_INF, u = 0.0f;
#pragma unroll
  for (int i = 0; i < 32; ++i) {
    float x  = s[base + i];
    float mf = fmaxf(m, x);
    u = u * __expf(m - mf) + __expf(x - mf);
    m = mf;
  }

  float sm = m, su = u;
#pragma unroll
  for (int off = 1; off < 32; off <<= 1) {
    float om = __shfl_up(sm, off, 32);
    float ou = __shfl_up(su, off, 32);
    if (lane >= off) lse_comb(sm, su, om, ou);
  }

  __shared__ float wm[8], wu[8], em[8], eu[8];
  if (lane == 31) { wm[wave] = sm; wu[wave] = su; }
  __syncthreads();
  if (tid == 0) {
    float am = NEG_INF, au = 0.0f;
    for (int w = 0; w < 8; ++w) {
      em[w] = am; eu[w] = au;
      lse_comb(am, au, wm[w], wu[w]);
    }
  }
  __syncthreads();

  float pm = em[wave], pu = eu[wave];
  float qm = __shfl_up(sm, 1, 32);
  float qu = __shfl_up(su, 1, 32);
  if (lane > 0) lse_comb(pm, pu, qm, qu);

  for (int i = 0; i < 32; ++i) {
    float x  = s[base + i];
    float mf = fmaxf(pm, x);
    pu = pu * __expf(pm - mf) + __expf(x - mf);
    pm = mf;
    Mout[base + i] = pm;
    Uout[base + i] = pu;
    iU[base + i]   = 1.0f / pu;
  }
}

// ---------------------------------------------------------------------------
// Kernel 2b: per-chunk WMMA A-operand tiles, pre-scaled by invU.
//   Tn[t,i] = exp(s_i - M_t) * invU_t  (i<=t within the 16-chunk, else 0)
// stored in the exact f32 16x16x4 A lane layout: slice j, lane -> {K=i0, K=i0+1},
// i0 = 4j + (lane>=16 ? 2 : 0), row = lane&15.
// Also CSn[t] = exp(M[chunkbase-1] - M_t) * invU_t (carry scale, normalized).
// ---------------------------------------------------------------------------
__global__ void aaren_tiles(const float* __restrict__ s,
                            const float* __restrict__ M,
                            const float* __restrict__ iU,
                            float* __restrict__ Tbuf,
                            float* __restrict__ CSn) {
  const int c    = blockIdx.x;        // chunk id, 0..NCHK-1
  const int t0   = c * 16;
  const int lane = threadIdx.x;       // 32
  const int nn   = lane & 15;
  const int hi   = (lane & 16) ? 1 : 0;

  const float sL = s[t0 + nn];
  const float mL = M[t0 + nn];
  const float uL = iU[t0 + nn];
  const float Min = (t0 == 0) ? NEG_INF : M[t0 - 1];

  if (lane < 16) CSn[t0 + lane] = __expf(Min - mL) * uL;

#pragma unroll
  for (int j = 0; j < 4; ++j) {
    const int i0 = 4 * j + (hi ? 2 : 0);
    const float si0 = __shfl(sL, i0, 32);
    const float si1 = __shfl(sL, i0 + 1, 32);
    v2f a;
    a.x = (i0     <= nn) ? __expf(si0 - mL) * uL : 0.0f;
    a.y = (i0 + 1 <= nn) ? __expf(si1 - mL) * uL : 0.0f;
    *reinterpret_cast<v2f*>(Tbuf + ((size_t)(c * 4 + j) * 32 + lane) * 2) = a;
  }
}

// ---------------------------------------------------------------------------
// Kernel 3a: per-group column sums Wg[g,d] = sum_{i in g} exp(s_i - M_end(g)) V[i,d]
// grid (DCH/32, GRP), 32 threads; lane owns one column -> 128B/row coalesced.
// This is the HBM pass over V.
// ---------------------------------------------------------------------------
__global__ void aaren_gsum(const float* __restrict__ V,
                           const float* __restrict__ s,
                           const float* __restrict__ M,
                           float* __restrict__ WGb) {
  const int lane = threadIdx.x;
  const int col  = blockIdx.x * 32 + lane;
  const int g    = blockIdx.y;
  const int t0   = g * RPG;
  const float mend = M[t0 + RPG - 1];
  float a0 = 0.f, a1 = 0.f, a2 = 0.f, a3 = 0.f;
#pragma unroll 2
  for (int r = 0; r < RPG; r += 4) {
    float w0 = __expf(s[t0 + r + 0] - mend);
    float w1 = __expf(s[t0 + r + 1] - mend);
    float w2 = __expf(s[t0 + r + 2] - mend);
    float w3 = __expf(s[t0 + r + 3] - mend);
    a0 = fmaf(w0, V[(size_t)(t0 + r + 0) * DCH + col], a0);
    a1 = fmaf(w1, V[(size_t)(t0 + r + 1) * DCH + col], a1);
    a2 = fmaf(w2, V[(size_t)(t0 + r + 2) * DCH + col], a2);
    a3 = fmaf(w3, V[(size_t)(t0 + r + 3) * DCH + col], a3);
  }
  WGb[(size_t)g * DCH + col] = (a0 + a1) + (a2 + a3);
}

// ---------------------------------------------------------------------------
// Kernel 3b: serial prefix over the GRP group sums, per column, in-place:
// after this, WGb[g,d] = carry-in numerator for group g at scale M_end(g-1).
// ---------------------------------------------------------------------------
__global__ void aaren_gscan(const float* __restrict__ M,
                            float* __restrict__ WGb) {
  const int d = blockIdx.x * blockDim.x + threadIdx.x;
  float n = 0.0f;
  float mprev = NEG_INF;
  for (int g = 0; g < GRP; ++g) {
    float mend = M[g * RPG + RPG - 1];
    float w = WGb[(size_t)g * DCH + d];
    WGb[(size_t)g * DCH + d] = n;            // carry-in (scale mprev)
    n = n * __expf(mprev - mend) + w;        // rescale & absorb group
    mprev = mend;
  }
}

// ---------------------------------------------------------------------------
// Kernel 3c: outputs via fp32 WMMA.  Per 16-row chunk:
//   out_rows = Tn @ Vtile + carry (x) CSn   (already normalized)
//   new carry = out_row15 * U[t0+15]        (un-normalize row 15)
// grid (DCH/32, GRP), 1 wave per block; two 16-col WMMA tiles per slab.
// V and Tbuf are L2-resident here.
// ---------------------------------------------------------------------------
__global__ void aaren_out(const float* __restrict__ V,
                          const float* __restrict__ U,
                          const float* __restrict__ CSn,
                          const float* __restrict__ Tbuf,
                          const float* __restrict__ WGb,
                          float* __restrict__ out) {
  const int lane = threadIdx.x;
  const int g    = blockIdx.y;
  const int col0 = blockIdx.x * 32;
  const int nn   = lane & 15;
  const int hi   = (lane & 16) ? 1 : 0;

  // carry for column col0+lane (un-normalized numerator at scale M[t0-1])
  float carry = WGb[(size_t)g * DCH + col0 + lane];

  for (int ch = 0; ch < RPG / 16; ++ch) {
    const int t0 = g * RPG + ch * 16;
    const int c  = g * (RPG / 16) + ch;      // global chunk id

    const float csnL = CSn[t0 + nn];         // lane holds row (lane&15) scale
    const float cN0  = __shfl(carry, nn, 32);
    const float cN1  = __shfl(carry, nn + 16, 32);

    v8f acc0, acc1;
#pragma unroll
    for (int r = 0; r < 8; ++r) {
      float sc = __shfl(csnL, r + (hi ? 8 : 0), 32);
      acc0[r] = cN0 * sc;
      acc1[r] = cN1 * sc;
    }

    const v2f* Ap = reinterpret_cast<const v2f*>(Tbuf + (size_t)c * 256);
#pragma unroll
    for (int j = 0; j < 4; ++j) {
      const v2f a = Ap[j * 32 + lane];       // pre-scaled triangular slice
      const int i0 = 4 * j + (hi ? 2 : 0);
      const float* vr0 = V + (size_t)(t0 + i0) * DCH + col0;
      const float* vr1 = V + (size_t)(t0 + i0 + 1) * DCH + col0;
      v2f b0, b1;
      b0.x = vr0[nn];      b0.y = vr1[nn];
      b1.x = vr0[nn + 16]; b1.y = vr1[nn + 16];
      acc0 = __builtin_amdgcn_wmma_f32_16x16x4_f32(false, a, false, b0,
                                                   (short)0, acc0, false, false);
      acc1 = __builtin_amdgcn_wmma_f32_16x16x4_f32(false, a, false, b1,
                                                   (short)0, acc1, false, false);
    }

    // acc is already the normalized output; store directly
#pragma unroll
    for (int r = 0; r < 8; ++r) {
      int mrow = r + (hi ? 8 : 0);
      size_t ro = (size_t)(t0 + mrow) * DCH + col0;
      out[ro + nn]      = acc0[r];
      out[ro + 16 + nn] = acc1[r];
    }

    // new carry = un-normalized row 15 (VGPR7, lanes 16..31)
    const float Uend = U[t0 + 15];
    float ca = __shfl(acc0[7], 16 + nn, 32);
    float cb = __shfl(acc1[7], 16 + nn, 32);
    carry = ((lane < 16) ? ca : cb) * Uend;
  }
}

// ---------------------------------------------------------------------------
extern "C" void kernel_launch(void* const* d_in, const int* in_sizes, int n_in,
                              void* d_out, int out_size, void* d_ws, size_t ws_size,
                              hipStream_t stream) {
  const float* K = (const float*)d_in[0];   // [L, D]
  const float* V = (const float*)d_in[1];   // [L, D]
  const float* q = (const float*)d_in[2];   // [D]
  float* out = (float*)d_out;               // [L, D]

  float* ws   = (float*)d_ws;
  float* s    = ws;                          // [L]
  float* M    = ws + LTOK;                   // [L]
  float* U    = ws + 2 * LTOK;               // [L]
  float* iU   = ws + 3 * LTOK;               // [L]
  float* CSn  = ws + 4 * LTOK;               // [L]
  float* WGb  = ws + 5 * LTOK;               // [GRP, D]
  float* Tbuf = WGb + (size_t)GRP * DCH;     // [NCHK * 256]

  aaren_logits<<<dim3(LTOK / 8), dim3(256), 0, stream>>>(K, q, s);
  aaren_scan<<<dim3(1), dim3(256), 0, stream>>>(s, M, U, iU);
  aaren_tiles<<<dim3(NCHK), dim3(32), 0, stream>>>(s, M, iU, Tbuf, CSn);
  aaren_gsum<<<dim3(DCH / 32, GRP), dim3(32), 0, stream>>>(V, s, M, WGb);
  aaren_gscan<<<dim3(DCH / 256), dim3(256), 0, stream>>>(M, WGb);
  aaren_out<<<dim3(DCH / 32, GRP), dim3(32), 0, stream>>>(V, U, CSn, Tbuf, WGb, out);
}